// MultiHeadAttention_35673998360864
// MI455X (gfx1250) — compile-verified
//
#include <hip/hip_runtime.h>

typedef float v2f __attribute__((ext_vector_type(2)));
typedef float v8f __attribute__((ext_vector_type(8)));

namespace {
constexpr int   kNHead  = 8;
constexpr int   kDk     = 64;
constexpr int   kLen    = 100;
constexpr int   kDModel = 512;
constexpr int   kBs     = 32;
constexpr float kTemp   = 20.0f;
constexpr float kEps    = 1e-12f;
constexpr int   kRows   = kBs * kLen;   // 3200 projected rows
constexpr int   kSStr   = 116;          // LDS stride (floats) for score matrix
}

__device__ __forceinline__ v8f wmma_f32(v2f a, v2f b, v8f c) {
  // V_WMMA_F32_16X16X4_F32 : D(16x16 f32) = A(16x4 f32) * B(4x16 f32) + C
  return __builtin_amdgcn_wmma_f32_16x16x4_f32(false, a, false, b, (short)0, c,
                                               false, false);
}

// ---------------------------------------------------------------------------
// Kernel 1: Y[R, c] = (X @ W^T)[R, c], then L2-normalize each (row, 64-col
// aligned group). Block = 4 waves = 16 rows x 64 cols = exactly 16 norm groups.
// grid (200, 8, 2): x = row tile, y = 64-col group, z = {q/w_qs -> qn, k/w_ks -> kn}
// ---------------------------------------------------------------------------
__global__ __launch_bounds__(128)
void proj_norm_kernel(const float* __restrict__ q, const float* __restrict__ k,
                      const float* __restrict__ w_qs, const float* __restrict__ w_ks,
                      float* __restrict__ qn, float* __restrict__ kn) {
  const float* X = (blockIdx.z == 0) ? q    : k;
  const float* W = (blockIdx.z == 0) ? w_qs : w_ks;
  float*       Y = (blockIdx.z == 0) ? qn   : kn;

  const int tid  = threadIdx.x;
  const int wave = tid >> 5;
  const int lane = tid & 31;
  const int r    = lane & 15;          // M (A) / N (B) index
  const int koff = (lane >> 4) * 2;    // K sub-offset {0,2}

  const int m0 = blockIdx.x * 16;
  const int n0 = blockIdx.y * 64 + wave * 16;

  const float* arow = X + (size_t)(m0 + r) * kDModel + koff;
  const float* brow = W + (size_t)(n0 + r) * kDModel + koff;  // B[k,n] = W[n,k]

  v8f acc = {};
#pragma unroll 4
  for (int kk = 0; kk < kDModel / 4; ++kk) {
    v2f a = *(const v2f*)(arow + 4 * kk);
    v2f b = *(const v2f*)(brow + 4 * kk);
    acc = wmma_f32(a, b, acc);
  }

  // Deterministic per-row sum-of-squares over this block's 64 columns.
  __shared__ float partial[4 * 16];    // [wave][row]
  const int rhalf = (lane >> 4) * 8;   // C layout: VGPR v holds M = v + rhalf
  float ss[8];
#pragma unroll
  for (int v = 0; v < 8; ++v) {
    float x = acc[v] * acc[v];
    x += __shfl_xor(x, 1);
    x += __shfl_xor(x, 2);
    x += __shfl_xor(x, 4);
    x += __shfl_xor(x, 8);             // all 16 lanes of the half now hold it
    ss[v] = x;
  }
  if ((lane & 15) == 0) {
#pragma unroll
    for (int v = 0; v < 8; ++v) partial[wave * 16 + rhalf + v] = ss[v];
  }
  __syncthreads();

#pragma unroll
  for (int v = 0; v < 8; ++v) {
    const int row = rhalf + v;
    const float s = partial[0 * 16 + row] + partial[1 * 16 + row] +
                    partial[2 * 16 + row] + partial[3 * 16 + row];
    const float scale = 1.0f / fmaxf(sqrtf(s), kEps);
    Y[(size_t)(m0 + row) * kDModel + n0 + r] = acc[v] * scale;
  }
}

// ---------------------------------------------------------------------------
// Kernel 2: per (qb, kb, h): S = Qn(100x64) Kn^T -> relu*20 -> softmax ->
// O = P(100x100) V(100x64), stored to out[kb, qb, l, h*64+d].
// grid (32, 32, 8), 8 waves. Qn/Kn/V stream from L2; S/P in LDS.
// ---------------------------------------------------------------------------
__global__ __launch_bounds__(256)
void attn_kernel(const float* __restrict__ qn, const float* __restrict__ kn,
                 const float* __restrict__ v, float* __restrict__ out) {
  const int qb = blockIdx.x, kb = blockIdx.y, h = blockIdx.z;
  const int tid  = threadIdx.x;
  const int wave = tid >> 5;
  const int lane = tid & 31;
  const int r    = lane & 15;
  const int koff = (lane >> 4) * 2;
  const int rhalf = (lane >> 4) * 8;

  __shared__ __align__(16) float Ss[112 * kSStr];   // ~51 KB

  const float* gq = qn + (size_t)(qb * kNHead + h) * (kLen * kDk);
  const float* gk = kn + (size_t)(kb * kNHead + h) * (kLen * kDk);
  const float* gv = v  + (size_t)(kb * kNHead + h) * (kLen * kDk);

  // ---- GEMM1: S(112x112 padded) = Qn * Kn^T, K = 64 -> 16 WMMA steps ----
  for (int t = wave; t < 49; t += 8) {
    const int m0 = (t / 7) * 16, n0 = (t % 7) * 16;
    const int ra = (m0 + r < kLen) ? (m0 + r) : (kLen - 1);  // clamp: no OOB
    const int rb = (n0 + r < kLen) ? (n0 + r) : (kLen - 1);
    const float* ap = gq + (size_t)ra * kDk + koff;
    const float* bp = gk + (size_t)rb * kDk + koff;
    v8f acc = {};
#pragma unroll
    for (int kk = 0; kk < 16; ++kk) {
      v2f a = *(const v2f*)(ap + 4 * kk);
      v2f b = *(const v2f*)(bp + 4 * kk);
      acc = wmma_f32(a, b, acc);
    }
#pragma unroll
    for (int vv = 0; vv < 8; ++vv)
      Ss[(m0 + rhalf + vv) * kSStr + n0 + r] = acc[vv];
  }
  __syncthreads();

  // ---- softmax over m for rows 0..99 (relu * TEMP first) ----
  for (int row = wave; row < kLen; row += 8) {
    float* srow = &Ss[row * kSStr];
    float x0 = fmaxf(srow[lane +  0], 0.f) * kTemp;
    float x1 = fmaxf(srow[lane + 32], 0.f) * kTemp;
    float x2 = fmaxf(srow[lane + 64], 0.f) * kTemp;
    float x3 = (lane < 4) ? fmaxf(srow[lane + 96], 0.f) * kTemp : -1e30f;
    float mx = fmaxf(fmaxf(x0, x1), fmaxf(x2, x3));
#pragma unroll
    for (int off = 16; off >= 1; off >>= 1) mx = fmaxf(mx, __shfl_xor(mx, off));
    float e0 = __expf(x0 - mx);
    float e1 = __expf(x1 - mx);
    float e2 = __expf(x2 - mx);
    float e3 = (lane < 4) ? __expf(x3 - mx) : 0.f;
    float s = e0 + e1 + e2 + e3;
#pragma unroll
    for (int off = 16; off >= 1; off >>= 1) s += __shfl_xor(s, off);
    const float inv = 1.0f / s;
    srow[lane +  0] = e0 * inv;
    srow[lane + 32] = e1 * inv;
    srow[lane + 64] = e2 * inv;
    if (lane < 4) srow[lane + 96] = e3 * inv;
  }
  __syncthreads();

  // ---- GEMM2: O(100x64) = P(100x100) * V(100x64), K = 100 -> 25 steps ----
  float* outp = out + ((size_t)(kb * kBs + qb)) * (kLen * kDModel) + h * kDk;
  for (int t = wave; t < 28; t += 8) {
    const int m0 = (t >> 2) * 16, n0 = (t & 3) * 16;
    const float* ap = &Ss[(m0 + r) * kSStr + koff];
    v8f acc = {};
#pragma unroll
    for (int kk = 0; kk < 25; ++kk) {
      v2f a = *(const v2f*)(ap + 4 * kk);
      const int kidx = 4 * kk + koff;           // <= 98, +1 <= 99: in range
      v2f b;
      b.x = gv[(size_t)(kidx + 0) * kDk + n0 + r];
      b.y = gv[(size_t)(kidx + 1) * kDk + n0 + r];
      acc = wmma_f32(a, b, acc);
    }
#pragma unroll
    for (int vv = 0; vv < 8; ++vv) {
      const int l = m0 + rhalf + vv;
      if (l < kLen) outp[(size_t)l * kDModel + n0 + r] = acc[vv];
    }
  }
}

// ---------------------------------------------------------------------------
extern "C" void kernel_launch(void* const* d_in, const int* in_sizes, int n_in,
                              void* d_out, int out_size, void* d_ws, size_t ws_size,
                              hipStream_t stream) {
  (void)in_sizes; (void)n_in; (void)out_size; (void)ws_size;
  const float* q    = (const float*)d_in[0];
  const float* k    = (const float*)d_in[1];
  const float* v    = (const float*)d_in[2];
  const float* w_qs = (const float*)d_in[3];
  const float* w_ks = (const float*)d_in[4];
  float* out = (float*)d_out;

  float* qn = (float*)d_ws;                       // 3200*512 floats
  float* kn = qn + (size_t)kRows * kDModel;       // 3200*512 floats

  dim3 g1(kRows / 16, kDModel / 64, 2);           // (200, 8, 2)
  proj_norm_kernel<<<g1, 128, 0, stream>>>(q, k, w_qs, w_ks, qn, kn);

  dim3 g2(kBs, kBs, kNHead);                      // (32, 32, 8)
  attn_kernel<<<g2, 256, 0, stream>>>(qn, kn, v, out);
}